// TrainableMixtureLowRankRNN_48369921687957
// MI455X (gfx1250) — compile-verified
//
#include <hip/hip_runtime.h>
#include <math.h>

// Problem constants (from reference)
#define HIDDEN 1024
#define DD     12       // 2*RANK + INPUT_SIZE
#define NMIX   4
#define INSZ   8
#define BATCH  64
#define SEQ    512
#define NOUT   10       // RANK + INPUT_SIZE (span columns)
#define ALPHA  0.1f
#define CJ     (500.0f/1024.0f)   // BASE_SCALE / HIDDEN_SIZE

typedef __attribute__((ext_vector_type(2))) float v2f;
typedef __attribute__((ext_vector_type(8))) float v8f;

// gfx1250 has a hardware tanh transcendental (v_tanh_f32).  Call the builtin
// unconditionally so compile feedback proves the hot loop uses it.
__device__ __forceinline__ float fast_tanh(float x) {
    return __builtin_amdgcn_tanhf(x);
}

// ---------------------------------------------------------------------------
// Kernel 1: mixed[h][e] = sum_k w[k] * ( means[k][e] + sum_d eps[k][h][d]*L[k][e][d] )
// L = tril(raw,-1) + (|diag-1e-12|+1e-12) on diagonal;  w = clip/normalize.
// 1024 threads total, one hidden row each.  (tiny)
// ---------------------------------------------------------------------------
__global__ void __launch_bounds__(256)
setup_mixed_kernel(const float* __restrict__ eps,
                   const float* __restrict__ means,
                   const float* __restrict__ str,
                   const float* __restrict__ mw,
                   float* __restrict__ mixed) {
    int h = blockIdx.x * blockDim.x + threadIdx.x;  // 0..1023
    float w[NMIX]; float wsum = 0.f;
    #pragma unroll
    for (int k = 0; k < NMIX; k++) {
        float v = mw[k]; v = (v < 1e-6f) ? 1e-6f : v;
        w[k] = v; wsum += v;
    }
    float winv = 1.0f / wsum;
    float acc[DD];
    #pragma unroll
    for (int e = 0; e < DD; e++) acc[e] = 0.f;
    for (int k = 0; k < NMIX; k++) {
        const float* ek = eps + ((size_t)k * HIDDEN + h) * DD;
        float ev[DD];
        #pragma unroll
        for (int d = 0; d < DD; d++) ev[d] = ek[d];
        float wk = w[k] * winv;
        #pragma unroll
        for (int e = 0; e < DD; e++) {
            float s = means[k * DD + e];
            for (int d = 0; d <= e; d++) {
                float raw = str[(k * DD + e) * DD + d];
                float l = (d == e) ? (fabsf(raw - 1e-12f) + 1e-12f) : raw;
                s += ev[d] * l;
            }
            acc[e] += wk * s;
        }
    }
    float* o = mixed + (size_t)h * DD;
    #pragma unroll
    for (int e = 0; e < DD; e++) o[e] = acc[e];
}

// ---------------------------------------------------------------------------
// Kernel 2: pinv of span (1024x10) via normal equations.
// span[h] = [mixed[h][0:2], mixed[h][4:12]].  G = span^T span (10x10, LDS
// float atomics), Gauss-Jordan invert on thread 0, then
// pinvT[h][n] = sum_e Ginv[n][e]*span[h][e], padded to 16 cols (n>=10 -> 0).
// One block of 256 threads.
// ---------------------------------------------------------------------------
__global__ void __launch_bounds__(256)
pinv_kernel(const float* __restrict__ mixed, float* __restrict__ pinvT) {
    __shared__ float G[100];
    __shared__ float Gi[100];
    int tid = threadIdx.x;
    if (tid < 100) G[tid] = 0.f;
    __syncthreads();

    float srow[4][NOUT];
    for (int q = 0; q < 4; q++) {
        int h = tid + 256 * q;
        const float* mr = mixed + (size_t)h * DD;
        srow[q][0] = mr[0]; srow[q][1] = mr[1];
        #pragma unroll
        for (int e = 0; e < 8; e++) srow[q][2 + e] = mr[4 + e];
        for (int a = 0; a < NOUT; a++)
            for (int b = a; b < NOUT; b++)
                atomicAdd(&G[a * 10 + b], srow[q][a] * srow[q][b]);
    }
    __syncthreads();

    if (tid == 0) {
        float A[10][20];
        for (int r = 0; r < 10; r++) {
            for (int c = 0; c < 10; c++)
                A[r][c] = (r <= c) ? G[r * 10 + c] : G[c * 10 + r];
            for (int c = 0; c < 10; c++)
                A[r][10 + c] = (r == c) ? 1.f : 0.f;
        }
        for (int p = 0; p < 10; p++) {
            int best = p; float bv = fabsf(A[p][p]);
            for (int r = p + 1; r < 10; r++) {
                float v = fabsf(A[r][p]);
                if (v > bv) { bv = v; best = r; }
            }
            if (best != p)
                for (int c = 0; c < 20; c++) {
                    float t = A[p][c]; A[p][c] = A[best][c]; A[best][c] = t;
                }
            float ip = 1.0f / A[p][p];
            for (int c = 0; c < 20; c++) A[p][c] *= ip;
            for (int r = 0; r < 10; r++) {
                if (r == p) continue;
                float f = A[r][p];
                for (int c = 0; c < 20; c++) A[r][c] -= f * A[p][c];
            }
        }
        for (int r = 0; r < 10; r++)
            for (int c = 0; c < 10; c++) Gi[r * 10 + c] = A[r][10 + c];
    }
    __syncthreads();

    for (int q = 0; q < 4; q++) {
        int h = tid + 256 * q;
        float* o = pinvT + (size_t)h * 16;
        #pragma unroll
        for (int n = 0; n < NOUT; n++) {
            float s = 0.f;
            #pragma unroll
            for (int e = 0; e < NOUT; e++) s += Gi[n * 10 + e] * srow[q][e];
            o[n] = s;
        }
        #pragma unroll
        for (int n = NOUT; n < 16; n++) o[n] = 0.f;
    }
}

// ---------------------------------------------------------------------------
// Kernel 3: fused sequential scan + WMMA output projection.
// One block per batch row (64 blocks x 256 threads, 8 waves).  h kept in
// registers (4 elems/thread).  Rank-2 recurrence: u_r = sum_j tanh(h_j) n_jr
// via wave shfl_xor + 1 LDS hop + 1 barrier per step.  Every 16 steps the
// 16x1024 h-chunk in LDS is projected through pinvT with
// V_WMMA_F32_16X16X4_F32: each wave owns a K=128 slice (B fragments are
// loop-invariant, preloaded to VGPRs), partial C tiles reduced through LDS.
// ---------------------------------------------------------------------------
__global__ void __launch_bounds__(256)
rnn_fused_kernel(const float* __restrict__ x,
                 const float* __restrict__ mixed,
                 const float* __restrict__ pinvT,
                 float* __restrict__ out) {
    __shared__ float hbuf[16][HIDDEN];      // 64 KB: 16 timesteps of h
    __shared__ float cpart[8][256];         // 8 KB: per-wave partial C tiles
    __shared__ float redbuf[2][8][2];       // double-buffered wave partials

    const int b    = blockIdx.x;
    const int tid  = threadIdx.x;
    const int wave = tid >> 5;
    const int lane = tid & 31;
    const int row  = lane & 15;             // M row (A) / N col (B)
    const int kadd = (lane >> 4) * 2;       // K sub-offset per lane half

    // Per-thread constants from mixed: m (cols 0:2), n (2:4), I (4:12)
    float m0[4], m1[4], n0[4], n1[4], Ic[4][INSZ];
    #pragma unroll
    for (int q = 0; q < 4; q++) {
        int i = tid + 256 * q;
        const float* mr = mixed + (size_t)i * DD;
        m0[q] = mr[0]; m1[q] = mr[1]; n0[q] = mr[2]; n1[q] = mr[3];
        #pragma unroll
        for (int k = 0; k < INSZ; k++) Ic[q][k] = mr[4 + k];
    }

    // Preload this wave's B fragments (pinvT rows k..k+3, col = lane&15).
    // B 4x16 layout: VGPR0 = K {0,2} by lane half, VGPR1 = K {1,3}.
    v2f Bf[32];
    #pragma unroll
    for (int j = 0; j < 32; j++) {
        int k0 = wave * 128 + 4 * j;
        Bf[j].x = pinvT[(size_t)(k0 + kadd)     * 16 + row];
        Bf[j].y = pinvT[(size_t)(k0 + kadd + 1) * 16 + row];
    }

    float h[4] = {0.f, 0.f, 0.f, 0.f};
    const float4* xb = reinterpret_cast<const float4*>(x + (size_t)b * SEQ * INSZ);

    for (int t = 0; t < SEQ; t++) {
        // --- rank-2 reduction: u_r = sum_j tanh(h_j) * n[j][r] ---
        float p0 = 0.f, p1 = 0.f;
        #pragma unroll
        for (int q = 0; q < 4; q++) {
            float th = fast_tanh(h[q]);
            p0 += th * n0[q];
            p1 += th * n1[q];
        }
        #pragma unroll
        for (int off = 16; off > 0; off >>= 1) {
            p0 += __shfl_xor(p0, off, 32);
            p1 += __shfl_xor(p1, off, 32);
        }
        if (lane == 0) { redbuf[t & 1][wave][0] = p0; redbuf[t & 1][wave][1] = p1; }
        __syncthreads();
        float u0 = 0.f, u1 = 0.f;
        #pragma unroll
        for (int w2 = 0; w2 < 8; w2++) {
            u0 += redbuf[t & 1][w2][0];
            u1 += redbuf[t & 1][w2][1];
        }

        // --- x_t @ I^T + state update ---
        float4 x0 = xb[t * 2 + 0];
        float4 x1 = xb[t * 2 + 1];
        float xv[8] = {x0.x, x0.y, x0.z, x0.w, x1.x, x1.y, x1.z, x1.w};
        #pragma unroll
        for (int q = 0; q < 4; q++) {
            float dx = 0.f;
            #pragma unroll
            for (int k = 0; k < INSZ; k++) dx += xv[k] * Ic[q][k];
            float hn = h[q] + ALPHA * (-h[q] + CJ * (u0 * m0[q] + u1 * m1[q]) + dx);
            h[q] = hn;
            hbuf[t & 15][tid + 256 * q] = hn;
        }

        // --- every 16 steps: WMMA projection of the 16x1024 chunk ---
        if ((t & 15) == 15) {
            __syncthreads();   // hbuf chunk complete
            v8f c = {0.f, 0.f, 0.f, 0.f, 0.f, 0.f, 0.f, 0.f};
            #pragma unroll
            for (int j = 0; j < 32; j++) {
                int k0 = wave * 128 + 4 * j;
                v2f a;
                a.x = hbuf[row][k0 + kadd];       // A 16x4 f32: lane=M, vgpr=K
                a.y = hbuf[row][k0 + kadd + 1];
                c = __builtin_amdgcn_wmma_f32_16x16x4_f32(
                        false, a, false, Bf[j], (short)0, c, false, false);
            }
            #pragma unroll
            for (int r = 0; r < 8; r++) cpart[wave][r * 32 + lane] = c[r];
            __syncthreads();   // partials ready; also fences hbuf reads
            int mm = tid >> 4, nn = tid & 15;
            if (nn < NOUT) {
                float s = 0.f;
                #pragma unroll
                for (int w2 = 0; w2 < 8; w2++)
                    s += cpart[w2][(mm & 7) * 32 + (mm >> 3) * 16 + nn];
                int tt = t - 15 + mm;
                out[((size_t)b * SEQ + tt) * NOUT + nn] = s;
            }
        }
    }
}

// ---------------------------------------------------------------------------
// d_in order: x, eps, means, scale_tril_raw, mixture_weights   (all f32)
// d_out: (64, 512, 10) f32.   d_ws: mixed (48KB) + pinvT (64KB).
// ---------------------------------------------------------------------------
extern "C" void kernel_launch(void* const* d_in, const int* in_sizes, int n_in,
                              void* d_out, int out_size, void* d_ws, size_t ws_size,
                              hipStream_t stream) {
    const float* x     = (const float*)d_in[0];
    const float* eps   = (const float*)d_in[1];
    const float* means = (const float*)d_in[2];
    const float* str   = (const float*)d_in[3];
    const float* mw    = (const float*)d_in[4];
    float* out = (float*)d_out;

    float* mixed = (float*)d_ws;                                   // 1024*12 f32
    float* pinvT = (float*)((char*)d_ws + HIDDEN * DD * sizeof(float)); // 1024*16 f32

    setup_mixed_kernel<<<4, 256, 0, stream>>>(eps, means, str, mw, mixed);
    pinv_kernel<<<1, 256, 0, stream>>>(mixed, pinvT);
    rnn_fused_kernel<<<BATCH, 256, 0, stream>>>(x, mixed, pinvT, out);
}